// PBFDKalmanFilter_76244259438923
// MI455X (gfx1250) — compile-verified
//
#include <hip/hip_runtime.h>
#include <hip/hip_bf16.h>
#include <math.h>

// ---------------------------------------------------------------------------
// PBFD Kalman feedback canceller, DFT-as-GEMM formulation for CDNA5 (gfx1250).
// All rfft/irfft ops become batched GEMMs executed with V_WMMA_F32_16X16X4_F32,
// fed from LDS (A and B tiles cooperatively staged, conflict-free stride-68).
// ---------------------------------------------------------------------------

#define FFTN   512
#define HOP    256
#define NBLK   4
#define NBIN   257
#define BATCH  4096
#define NF     576      // padded freq layout width: Re @ [0..256], Im @ [288..544]
#define IMOFF  288
#define SFAC   0.99999f
#define ALPHAE 0.7f

#define KC     64       // K chunk staged in LDS
#define LSTR   68       // LDS row stride (floats): banks (4*row + k) % 64 -> conflict-free

typedef float v2f __attribute__((ext_vector_type(2)));
typedef float v8f __attribute__((ext_vector_type(8)));

// ---------------- workspace layout (float offsets) -------------------------
#define OFF_WF512   0u                                    // 512 x NF fwd rfft (frames)
#define OFF_WF256P  (OFF_WF512  + 512u*NF)                // 256 x NF fwd rfft, zero-front-pad phase
#define OFF_WF256   (OFF_WF256P + 256u*NF)                // 256 x NF fwd rfft, head samples
#define OFF_WINVT   (OFF_WF256  + 256u*NF)                // NF x 256 irfft -> tail 256
#define OFF_WINVH   (OFF_WINVT  + (unsigned)NF*HOP)       // NF x 256 irfft -> head 256
#define OFF_XB      (OFF_WINVH  + (unsigned)NF*HOP)       // 16384 x NF   X spectra
#define OFF_KE      (OFF_XB + 16384u*NF)                  // 16384 x NF   K*E  (later fd_dH)
#define OFF_MF      (OFF_KE + 16384u*NF)                  // 4096 x NF    fd_mic
#define OFF_FDEST   (OFF_MF + 4096u*NF)                   // 4096 x NF    fd_est_fb
#define OFF_DH      OFF_MF                                // 16384 x 256  dH head (aliases MF+FDEST, both dead)
#define OFF_FB      (OFF_FDEST + 4096u*NF)                // 4096 x NF    fd_fb_hat / fd_fb_hat2
#define OFF_EST     (OFF_FB + 4096u*NF)                   // 4096 x 256   est_fb / est_fb2
// total = OFF_EST + 4096*256 = 27,885,568 floats (~106.4 MiB)

// ---------------- weight generation ----------------------------------------
__global__ void k_fwd_weights(float* __restrict__ W, int nrows, int alt) {
  int idx = blockIdx.x * blockDim.x + threadIdx.x;
  if (idx >= nrows * NF) return;
  int n = idx / NF, c = idx % NF;
  float val = 0.0f;
  int k = -1; bool im = false;
  if (c <= 256)                          { k = c; }
  else if (c >= IMOFF && c <= IMOFF+256) { k = c - IMOFF; im = true; }
  if (k >= 0) {
    float th = (float)((n * k) & (FFTN - 1)) * (6.283185307179586f / FFTN);
    float s  = (alt && (k & 1)) ? -1.0f : 1.0f;   // (-1)^k from 256-sample front zero-pad
    val = im ? (-s * sinf(th)) : (s * cosf(th));
  }
  W[idx] = val;
}

// irfft basis (1/512 folded in). tail=1 -> samples n=256..511, tail=0 -> n=0..255.
__global__ void k_inv_weights(float* __restrict__ W, int tail) {
  int idx = blockIdx.x * blockDim.x + threadIdx.x;
  if (idx >= NF * HOP) return;
  int j = idx / HOP, m = idx % HOP;
  const float inv = 1.0f / (float)FFTN;
  float val = 0.0f;
  if (j <= 256) {                       // Re(C_k) basis
    int k = j;
    if (k == 0)        val = inv;
    else if (k == 256) val = (m & 1) ? -inv : inv;           // (-1)^n, n = m (+256 for tail)
    else {
      float th = (float)((m * k) & (FFTN - 1)) * (6.283185307179586f / FFTN);
      float s  = (tail && (k & 1)) ? -1.0f : 1.0f;           // cos(pi k + th) = (-1)^k cos th
      val = 2.0f * inv * s * cosf(th);
    }
  } else if (j >= IMOFF && j <= IMOFF + 256) {  // Im(C_k) basis (k=0,256 ignored per irfft)
    int k = j - IMOFF;
    if (k >= 1 && k <= 255) {
      float th = (float)((m * k) & (FFTN - 1)) * (6.283185307179586f / FFTN);
      float s  = (tail && (k & 1)) ? -1.0f : 1.0f;
      val = -2.0f * inv * s * sinf(th);
    }
  }
  W[idx] = val;
}

// ---------------- WMMA GEMM: D[M x N] = A[M x K] * B[K x N] ----------------
// 256 threads = 8 waves; block tile 128(M) x 64(N); wave tile 16 x 64.
// K chunked by KC=64; A (128xKC) and B (KCx64, transposed) staged in LDS.
// FRAMES=true: row r of A = loudspeaker_buffer[r/4 * 1280 + (r%4)*256 ..+512]
template<bool FRAMES>
__global__ __launch_bounds__(256)
void k_gemm_wmma(const float* __restrict__ A, int lda,
                 const float* __restrict__ B, int ldb,
                 float* __restrict__ D, int ldd, int K)
{
  __shared__ float lds_a[128 * LSTR];   // [row 0..127][k 0..63], stride 68
  __shared__ float lds_b[64 * LSTR];    // [n 0..63][k 0..63], transposed, stride 68

  const int t    = threadIdx.x;
  const int lane = t & 31;
  const int wave = t >> 5;
  const int half = lane >> 4;     // 0: lanes 0-15, 1: lanes 16-31
  const int l16  = lane & 15;
  const int mblk = blockIdx.y * 128;
  const int n0   = blockIdx.x * 64;

  const int sa_col = 2 * lane;    // A staging: wave w stages rows w+8i, cols 2*lane..+1
  const int sb_n   = t & 63;      // B staging: col n = t&63, rows k = (t>>6)*16 + i
  const int sb_k0  = (t >> 6) * 16;

  v8f acc0 = {}, acc1 = {}, acc2 = {}, acc3 = {};

  for (int kc = 0; kc < K; kc += KC) {
    // ---- stage A tile: coalesced 256B row reads, conflict-free ds_store_b64
    #pragma unroll
    for (int i = 0; i < 16; ++i) {
      int r = wave + 8 * i;                    // relative row 0..127
      int R = mblk + r;
      const float* src = FRAMES
        ? (A + (size_t)(R >> 2) * 1280 + (size_t)(R & 3) * HOP + kc + sa_col)
        : (A + (size_t)R * lda + kc + sa_col);
      v2f v; v.x = src[0]; v.y = src[1];
      *(v2f*)&lds_a[r * LSTR + sa_col] = v;
    }
    // ---- stage B tile transposed: lds_b[n][k] = B[kc+k][n0+n]
    #pragma unroll
    for (int i = 0; i < 16; ++i) {
      int k = sb_k0 + i;
      lds_b[sb_n * LSTR + k] = B[(size_t)(kc + k) * ldb + n0 + sb_n];
    }
    __syncthreads();

    // ---- 16 WMMA k-steps from LDS: 5x ds_load_b64 + 4x wmma per step
    #pragma unroll
    for (int s = 0; s < 16; ++s) {
      const int ka = 4 * s + 2 * half;         // A/B VGPR pair holds K = ka, ka+1
      v2f a  = *(const v2f*)&lds_a[(wave * 16 + l16) * LSTR + ka];
      v2f b0 = *(const v2f*)&lds_b[(l16 +  0) * LSTR + ka];
      v2f b1 = *(const v2f*)&lds_b[(l16 + 16) * LSTR + ka];
      v2f b2 = *(const v2f*)&lds_b[(l16 + 32) * LSTR + ka];
      v2f b3 = *(const v2f*)&lds_b[(l16 + 48) * LSTR + ka];
      acc0 = __builtin_amdgcn_wmma_f32_16x16x4_f32(false, a, false, b0, (short)0, acc0, false, false);
      acc1 = __builtin_amdgcn_wmma_f32_16x16x4_f32(false, a, false, b1, (short)0, acc1, false, false);
      acc2 = __builtin_amdgcn_wmma_f32_16x16x4_f32(false, a, false, b2, (short)0, acc2, false, false);
      acc3 = __builtin_amdgcn_wmma_f32_16x16x4_f32(false, a, false, b3, (short)0, acc3, false, false);
    }
    __syncthreads();
  }

  // C/D layout: VGPR v -> row m0 + v + 8*half, col n0 + l16 (+16 per tile)
  const int m0 = mblk + wave * 16;
  float* __restrict__ dbase = D + (size_t)(m0 + 8 * half) * ldd + n0 + l16;
  #pragma unroll
  for (int v = 0; v < 8; ++v) {
    dbase[(size_t)v * ldd +  0] = acc0[v];
    dbase[(size_t)v * ldd + 16] = acc1[v];
    dbase[(size_t)v * ldd + 32] = acc2[v];
    dbase[(size_t)v * ldd + 48] = acc3[v];
  }
}

// ---------------- elementwise stages ---------------------------------------
// E1: fd_fb_hat[b,k] = sum_blk X[b,blk,k] * H[b,k,blk]     -> FB (4096 x NF)
__global__ __launch_bounds__(192)
void k_e1(const float* __restrict__ XB, const float* __restrict__ Hr,
          const float* __restrict__ Hi, float* __restrict__ FB)
{
  int col = blockIdx.x * 192 + threadIdx.x;   // 0..575
  int b   = blockIdx.y;
  float acc = 0.0f;
  int k = -1; bool im = false;
  if (col <= 256) k = col;
  else if (col >= IMOFF && col <= IMOFF + 256) { k = col - IMOFF; im = true; }
  if (k >= 0) {
    #pragma unroll
    for (int blk = 0; blk < NBLK; ++blk) {
      size_t xr = (size_t)(b * NBLK + blk) * NF + k;
      float Xr = XB[xr], Xi = XB[xr + IMOFF];
      size_t hidx = ((size_t)b * NBIN + k) * NBLK + blk;
      float hr = Hr[hidx], hi = Hi[hidx];
      acc += im ? (Xr * hi + Xi * hr) : (Xr * hr - Xi * hi);
    }
  }
  FB[(size_t)b * NF + col] = acc;
}

// E2: E = fd_mic - fd_est ; phi_e' ; R ; K = P1*conj(X)/R ; KE = K*E  -> KE (16384 x NF)
__global__ __launch_bounds__(192)
void k_e2(const float* __restrict__ MF, const float* __restrict__ FDEST,
          const float* __restrict__ XB, const float* __restrict__ phi_e,
          const float* __restrict__ P1, float* __restrict__ KE)
{
  int col = blockIdx.x * 192 + threadIdx.x;
  int r   = blockIdx.y;            // b*4 + blk
  int b   = r >> 2;
  float out = 0.0f;
  int k = -1; bool im = false;
  if (col <= 256) k = col;
  else if (col >= IMOFF && col <= IMOFF + 256) { k = col - IMOFF; im = true; }
  if (k >= 0) {
    size_t fb = (size_t)b * NF + k;
    float Ere = MF[fb]         - FDEST[fb];
    float Eim = MF[fb + IMOFF] - FDEST[fb + IMOFF];
    float pe  = ALPHAE * phi_e[(size_t)b * NBIN + k]
              + (1.0f - ALPHAE) * (Ere * Ere + Eim * Eim);
    size_t xr = (size_t)r * NF + k;
    float Xr = XB[xr], Xi = XB[xr + IMOFF];
    float p1 = P1[((size_t)b * NBIN + k) * NBLK + (r & 3)];
    float R  = (Xr * Xr + Xi * Xi) * p1 + 2.0f * pe + 1e-10f;
    float Kre =  p1 * Xr / R;
    float Kim = -p1 * Xi / R;
    out = im ? (Kre * Eim + Kim * Ere) : (Kre * Ere - Kim * Eim);
  }
  KE[(size_t)r * NF + col] = out;
}

// E3: H' = SF*(H + fd_dH) ; fd_fb_hat2[b,k] = sum_blk X*H'   -> FB2 (4096 x NF)
__global__ __launch_bounds__(192)
void k_e3(const float* __restrict__ XB, const float* __restrict__ Hr,
          const float* __restrict__ Hi, const float* __restrict__ FDDH,
          float* __restrict__ FB2)
{
  int col = blockIdx.x * 192 + threadIdx.x;
  int b   = blockIdx.y;
  float acc = 0.0f;
  int k = -1; bool im = false;
  if (col <= 256) k = col;
  else if (col >= IMOFF && col <= IMOFF + 256) { k = col - IMOFF; im = true; }
  if (k >= 0) {
    #pragma unroll
    for (int blk = 0; blk < NBLK; ++blk) {
      int r = b * NBLK + blk;
      size_t dh = (size_t)r * NF + k;
      size_t hidx = ((size_t)b * NBIN + k) * NBLK + blk;
      float Hnr = SFAC * (Hr[hidx] + FDDH[dh]);
      float Hni = SFAC * (Hi[hidx] + FDDH[dh + IMOFF]);
      float Xr = XB[dh], Xi = XB[dh + IMOFF];
      acc += im ? (Xr * Hni + Xi * Hnr) : (Xr * Hnr - Xi * Hni);
    }
  }
  FB2[(size_t)b * NF + col] = acc;
}

// E4: enhanced = mic - est_fb2
__global__ __launch_bounds__(256)
void k_e4(const float* __restrict__ mic, const float* __restrict__ EST,
          float* __restrict__ out)
{
  int i = blockIdx.x * 256 + threadIdx.x;   // 0 .. BATCH*HOP-1
  out[i] = mic[i] - EST[i];
}

// ---------------- launcher --------------------------------------------------
extern "C" void kernel_launch(void* const* d_in, const int* in_sizes, int n_in,
                              void* d_out, int out_size, void* d_ws, size_t ws_size,
                              hipStream_t stream)
{
  const float* mic   = (const float*)d_in[0];   // (4096, 256, 1)
  const float* lsb   = (const float*)d_in[1];   // (4096, 1280)
  const float* phi_e = (const float*)d_in[2];   // (4096, 257, 1)
  // d_in[3] phi_f: dead code in reference (only feeds unused P1_new)
  const float* P1    = (const float*)d_in[4];   // (4096, 257, 4, 1)
  const float* Hr    = (const float*)d_in[5];   // (4096, 257, 4, 1)
  const float* Hi    = (const float*)d_in[6];
  float* out = (float*)d_out;                   // (4096, 256, 1)
  float* W   = (float*)d_ws;

  float* Wf512 = W + OFF_WF512;
  float* Wf256p= W + OFF_WF256P;
  float* Wf256 = W + OFF_WF256;
  float* WinvT = W + OFF_WINVT;
  float* WinvH = W + OFF_WINVH;
  float* XB    = W + OFF_XB;
  float* KE    = W + OFF_KE;     // later re-used as fd_dH
  float* MF    = W + OFF_MF;
  float* FDEST = W + OFF_FDEST;
  float* DH    = W + OFF_DH;     // aliases MF/FDEST (dead by then)
  float* FB    = W + OFF_FB;     // fd_fb_hat, later fd_fb_hat2
  float* EST   = W + OFF_EST;    // est_fb, later est_fb2

  // weights (deterministic, regenerated every launch; graph-capture safe)
  k_fwd_weights<<<dim3((512u*NF + 255)/256), 256, 0, stream>>>(Wf512, 512, 0);
  k_fwd_weights<<<dim3((256u*NF + 255)/256), 256, 0, stream>>>(Wf256p, 256, 1);
  k_fwd_weights<<<dim3((256u*NF + 255)/256), 256, 0, stream>>>(Wf256, 256, 0);
  k_inv_weights<<<dim3(((unsigned)NF*HOP + 255)/256), 256, 0, stream>>>(WinvT, 1);
  k_inv_weights<<<dim3(((unsigned)NF*HOP + 255)/256), 256, 0, stream>>>(WinvH, 0);

  // G1: X = rfft(frames)                  (16384 x 512) x (512 x NF)
  k_gemm_wmma<true ><<<dim3(NF/64, 16384/128), 256, 0, stream>>>(lsb, 0, Wf512, NF, XB, NF, 512);
  // G2: fd_mic = rfft([0;mic])            (4096 x 256) x (256 x NF)
  k_gemm_wmma<false><<<dim3(NF/64,  4096/128), 256, 0, stream>>>(mic, HOP, Wf256p, NF, MF, NF, HOP);
  // E1: fd_fb_hat = sum_blk X*H
  k_e1<<<dim3(NF/192, BATCH), 192, 0, stream>>>(XB, Hr, Hi, FB);
  // G3: est_fb = irfft(fd_fb_hat)[256:]   (4096 x NF) x (NF x 256)
  k_gemm_wmma<false><<<dim3(HOP/64, 4096/128), 256, 0, stream>>>(FB, NF, WinvT, HOP, EST, HOP, NF);
  // G4: fd_est = rfft([0;est_fb])
  k_gemm_wmma<false><<<dim3(NF/64,  4096/128), 256, 0, stream>>>(EST, HOP, Wf256p, NF, FDEST, NF, HOP);
  // E2: Kalman gain, KE = K*E
  k_e2<<<dim3(NF/192, BATCH*NBLK), 192, 0, stream>>>(MF, FDEST, XB, phi_e, P1, KE);
  // G5: dH_head = irfft(K*E)[:256]        (16384 x NF) x (NF x 256)
  k_gemm_wmma<false><<<dim3(HOP/64, 16384/128), 256, 0, stream>>>(KE, NF, WinvH, HOP, DH, HOP, NF);
  // G6: fd_dH = rfft(dH_head)             (16384 x 256) x (256 x NF)  -> reuse KE buffer
  k_gemm_wmma<false><<<dim3(NF/64, 16384/128), 256, 0, stream>>>(DH, HOP, Wf256, NF, KE, NF, HOP);
  // E3: H' = SF*(H+fd_dH); fd_fb_hat2 = sum X*H'
  k_e3<<<dim3(NF/192, BATCH), 192, 0, stream>>>(XB, Hr, Hi, KE, FB);
  // G7: est_fb2 = irfft(fd_fb_hat2)[256:]
  k_gemm_wmma<false><<<dim3(HOP/64, 4096/128), 256, 0, stream>>>(FB, NF, WinvT, HOP, EST, HOP, NF);
  // E4: enhanced = mic - est_fb2
  k_e4<<<dim3(BATCH*HOP/256), 256, 0, stream>>>(mic, EST, out);
}